// LinearAttention_42623255446251
// MI455X (gfx1250) — compile-verified
//
#include <hip/hip_runtime.h>
#include <hip/hip_bf16.h>

// LinearAttention on MI455X (gfx1250, wave32).
// Pipeline: qglobal -> scores -> softmax(alpha) -> KV/Ksum (WMMA f32, split-K)
//           -> partial reduce -> out GEMM (WMMA f32) with fused denominator.

#define B_  4
#define H_  16
#define S_  4096
#define D_  64
#define BH_ 64
#define SPLIT_ 8          // split-K parts for the KV reduction GEMM
#define SROWS_ (S_/SPLIT_) // 512 rows per part
#define CH_ 64            // LDS chunk rows
#define LP_ 68            // padded LDS row stride (floats, multiple of 4)

typedef __attribute__((ext_vector_type(2))) float v2f;
typedef __attribute__((ext_vector_type(8))) float v8f;

// ---------------- Kernel 1: q_global = mean(Q over S) / sqrt(D) ----------------
__global__ void k_qglobal(const float* __restrict__ Q, float* __restrict__ qg) {
    int bh  = blockIdx.x;          // 64 blocks
    int tid = threadIdx.x;         // 256 threads
    int d = tid & 63, g = tid >> 6;
    const float* q = Q + (size_t)bh * S_ * D_;
    float acc = 0.f;
    for (int s = g; s < S_; s += 4) acc += q[(size_t)s * D_ + d];
    __shared__ float red[4][64];
    red[g][d] = acc;
    __syncthreads();
    if (tid < 64) {
        float v = red[0][tid] + red[1][tid] + red[2][tid] + red[3][tid];
        qg[bh * 64 + tid] = v * (1.0f / 32768.0f);   // (1/4096) * (1/8)
    }
}

// ---------------- Kernel 2a: scores[b,h,s] = qg . phi_K[s], partial max ----------------
__global__ void k_scores(const float* __restrict__ phiK, const float* __restrict__ qg,
                         float* __restrict__ scores, float* __restrict__ pmax) {
    int bh = blockIdx.x >> 3, part = blockIdx.x & 7;   // 512 blocks
    int tid = threadIdx.x;                             // 256 threads
    __shared__ __align__(16) float qgl[64];
    __shared__ float red[256];
    if (tid < 64) qgl[tid] = qg[bh * 64 + tid];
    __syncthreads();
    const float* pk = phiK + (size_t)bh * S_ * D_;
    const float4* qv = (const float4*)qgl;
    float lmax = -3.4e38f;
    for (int i = 0; i < 2; i++) {
        int s = part * 512 + i * 256 + tid;
        const float4* row = (const float4*)(pk + (size_t)s * D_);
        float sc = 0.f;
        for (int j = 0; j < 16; j++) {
            float4 a = row[j]; float4 b = qv[j];
            sc += a.x*b.x + a.y*b.y + a.z*b.z + a.w*b.w;
        }
        scores[bh * S_ + s] = sc;
        lmax = fmaxf(lmax, sc);
    }
    red[tid] = lmax;
    __syncthreads();
    for (int off = 128; off > 0; off >>= 1) {
        if (tid < off) red[tid] = fmaxf(red[tid], red[tid + off]);
        __syncthreads();
    }
    if (tid == 0) pmax[bh * 8 + part] = red[0];
}

// ---------------- Kernel 2b: alpha = softmax(scores) * S ----------------
__global__ void k_alpha(const float* __restrict__ scores, const float* __restrict__ pmax,
                        float* __restrict__ alpha) {
    int bh = blockIdx.x;           // 64 blocks
    int tid = threadIdx.x;         // 256 threads
    __shared__ float red[256];
    red[tid] = (tid < 8) ? pmax[bh * 8 + tid] : -3.4e38f;
    __syncthreads();
    for (int off = 128; off > 0; off >>= 1) {
        if (tid < off) red[tid] = fmaxf(red[tid], red[tid + off]);
        __syncthreads();
    }
    float mx = red[0];
    __syncthreads();
    const float* sc = scores + (size_t)bh * S_;
    float lsum = 0.f;
    for (int s = tid; s < S_; s += 256) lsum += __expf(sc[s] - mx);
    red[tid] = lsum;
    __syncthreads();
    for (int off = 128; off > 0; off >>= 1) {
        if (tid < off) red[tid] += red[tid + off];
        __syncthreads();
    }
    float scale = (float)S_ / red[0];
    for (int s = tid; s < S_; s += 256)
        alpha[bh * S_ + s] = __expf(sc[s] - mx) * scale;
}

// ---------------- Kernel 3: split-K  KV_part = (alpha*phi_K)^T @ V, Ksum_part ----------------
// 512 blocks (bh x 8 parts), 512 threads = 16 waves; wave (ti,tj) owns 16x16 tile of KV.
__global__ void k_kv(const float* __restrict__ phiK, const float* __restrict__ V,
                     const float* __restrict__ alpha, float* __restrict__ kvp,
                     float* __restrict__ ksump) {
    int bh = blockIdx.x >> 3, part = blockIdx.x & 7;
    int tid = threadIdx.x;     // 512
    __shared__ __align__(16) float pkl[CH_][LP_];
    __shared__ __align__(16) float vl[CH_][LP_];
    __shared__ float al[CH_];
    __shared__ float ksred[8][64];

    const float* pk = phiK  + (size_t)bh * S_ * D_ + (size_t)part * SROWS_ * D_;
    const float* vv = V     + (size_t)bh * S_ * D_ + (size_t)part * SROWS_ * D_;
    const float* ap = alpha + (size_t)bh * S_       + part * SROWS_;

    int w = tid >> 5, lane = tid & 31;
    int ti = w >> 2, tj = w & 3;
    int ll = lane & 15, hi = lane >> 4;
    int kd = tid & 63, kq = tid >> 6;     // Ksum: column d, 8-row group

    float ksacc = 0.f;
    v8f acc = {};

    for (int c = 0; c < SROWS_ / CH_; c++) {
        __syncthreads();
        const float4* src_pk = (const float4*)(pk + (size_t)c * CH_ * D_);
        const float4* src_v  = (const float4*)(vv + (size_t)c * CH_ * D_);
        for (int t = 0; t < 2; t++) {
            int idx = tid + t * 512;            // 1024 float4 per buffer
            int r = idx >> 4, c4 = idx & 15;
            *(float4*)&pkl[r][c4 * 4] = src_pk[idx];
            *(float4*)&vl[r][c4 * 4]  = src_v[idx];
        }
        if (tid < CH_) al[tid] = ap[c * CH_ + tid];
        __syncthreads();

        // Ksum partial: rows kq*8 .. kq*8+7, column kd
        for (int r = 0; r < 8; r++) {
            int rr = kq * 8 + r;
            ksacc += al[rr] * pkl[rr][kd];
        }

        // WMMA: A[d][k] = alpha[k]*phiK[k][d], B[k][n] = V[k][n]
        for (int kk = 0; kk < CH_; kk += 4) {
            int k0 = kk + 2 * hi;
            v2f a, b;
            a.x = al[k0]     * pkl[k0][16 * ti + ll];
            a.y = al[k0 + 1] * pkl[k0 + 1][16 * ti + ll];
            b.x = vl[k0][16 * tj + ll];
            b.y = vl[k0 + 1][16 * tj + ll];
            acc = __builtin_amdgcn_wmma_f32_16x16x4_f32(
                false, a, false, b, (short)0, acc, false, false);
        }
    }

    ksred[kq][kd] = ksacc;
    __syncthreads();
    if (tid < 64) {
        float s = 0.f;
        for (int q = 0; q < 8; q++) s += ksred[q][tid];
        ksump[(size_t)(bh * 8 + part) * 64 + tid] = s;
    }

    float* dst = kvp + (size_t)(bh * 8 + part) * (D_ * D_);
    for (int r = 0; r < 8; r++) {
        int m = 16 * ti + r + 8 * hi;
        dst[m * D_ + 16 * tj + ll] = acc[r];
    }
}

// ---------------- Kernel 3b: fixed-order reduce of split-K partials ----------------
__global__ void k_reduce(const float* __restrict__ kvp, const float* __restrict__ ksump,
                         float* __restrict__ kv, float* __restrict__ ksum) {
    int i = blockIdx.x * blockDim.x + threadIdx.x;   // 1024 x 256
    if (i < BH_ * D_ * D_) {
        int bh = i >> 12, e = i & 4095;
        float s = 0.f;
        for (int p = 0; p < SPLIT_; p++) s += kvp[(size_t)(bh * 8 + p) * 4096 + e];
        kv[i] = s;
    }
    if (i < BH_ * D_) {
        int bh = i >> 6, d = i & 63;
        float s = 0.f;
        for (int p = 0; p < SPLIT_; p++) s += ksump[(size_t)(bh * 8 + p) * 64 + d];
        ksum[i] = s;
    }
}

// ---------------- Kernel 4: out = (phi_Q @ KV) / (phi_Q . Ksum + 1e-6) ----------------
// 4096 blocks (bh x 64 row-chunks), 512 threads = 16 waves; wave owns a 16x16 out tile.
__global__ void k_out(const float* __restrict__ phiQ, const float* __restrict__ kv,
                      const float* __restrict__ ksum, float* __restrict__ out) {
    int bh = blockIdx.x >> 6;
    int row0 = (blockIdx.x & 63) * 64;
    int tid = threadIdx.x;
    __shared__ __align__(16) float kvl[D_][LP_];
    __shared__ float pqt[D_][LP_];   // transposed phi_Q chunk: pqt[k][row]
    __shared__ float ksl[64];
    __shared__ float den[64];

    const float* pq = phiQ + (size_t)bh * S_ * D_ + (size_t)row0 * D_;
    const float4* src_kv = (const float4*)(kv + (size_t)bh * (D_ * D_));
    const float4* src_pq = (const float4*)pq;
    for (int t = 0; t < 2; t++) {
        int idx = tid + t * 512;
        int r = idx >> 4, c4 = idx & 15;
        *(float4*)&kvl[r][c4 * 4] = src_kv[idx];
        float4 p = src_pq[idx];
        pqt[c4 * 4 + 0][r] = p.x;
        pqt[c4 * 4 + 1][r] = p.y;
        pqt[c4 * 4 + 2][r] = p.z;
        pqt[c4 * 4 + 3][r] = p.w;
    }
    if (tid < 64) ksl[tid] = ksum[bh * 64 + tid];
    __syncthreads();
    if (tid < 64) {
        float s = 0.f;
        for (int k = 0; k < 64; k++) s += pqt[k][tid] * ksl[k];
        den[tid] = s + 1e-6f;
    }
    __syncthreads();

    int w = tid >> 5, lane = tid & 31;
    int ti = w >> 2, tj = w & 3;
    int ll = lane & 15, hi = lane >> 4;
    v8f acc = {};
    for (int k = 0; k < 64; k += 4) {
        int k0 = k + 2 * hi;
        v2f a, b;
        a.x = pqt[k0][16 * ti + ll];
        a.y = pqt[k0 + 1][16 * ti + ll];
        b.x = kvl[k0][16 * tj + ll];
        b.y = kvl[k0 + 1][16 * tj + ll];
        acc = __builtin_amdgcn_wmma_f32_16x16x4_f32(
            false, a, false, b, (short)0, acc, false, false);
    }

    float* o = out + (size_t)bh * S_ * D_ + (size_t)row0 * D_;
    for (int r = 0; r < 8; r++) {
        int m = 16 * ti + r + 8 * hi;
        o[m * D_ + 16 * tj + ll] = acc[r] / den[m];
    }
}

extern "C" void kernel_launch(void* const* d_in, const int* in_sizes, int n_in,
                              void* d_out, int out_size, void* d_ws, size_t ws_size,
                              hipStream_t stream) {
    const float* Q    = (const float*)d_in[0];
    // d_in[1] = K is unused by the reference
    const float* V    = (const float*)d_in[2];
    const float* phiQ = (const float*)d_in[3];
    const float* phiK = (const float*)d_in[4];
    float* out = (float*)d_out;

    float* ws     = (float*)d_ws;
    float* qg     = ws;                       // BH*D            = 4096
    float* scores = qg + 4096;                // BH*S            = 262144
    float* pmax   = scores + 262144;          // BH*8            = 512
    float* alpha  = pmax + 512;               // BH*S            = 262144
    float* kvp    = alpha + 262144;           // 8*BH*D*D        = 2097152
    float* ksump  = kvp + 2097152;            // 8*BH*D          = 32768
    float* kv     = ksump + 32768;            // BH*D*D          = 262144
    float* ksum   = kv + 262144;              // BH*D            = 4096
    // total ~2.93M floats (~11.7 MB) of workspace

    k_qglobal<<<BH_, 256, 0, stream>>>(Q, qg);
    k_scores <<<BH_ * 8, 256, 0, stream>>>(phiK, qg, scores, pmax);
    k_alpha  <<<BH_, 256, 0, stream>>>(scores, pmax, alpha);
    k_kv     <<<BH_ * SPLIT_, 512, 0, stream>>>(phiK, V, alpha, kvp, ksump);
    k_reduce <<<1024, 256, 0, stream>>>(kvp, ksump, kv, ksum);
    k_out    <<<BH_ * (S_ / 64), 512, 0, stream>>>(phiQ, kv, ksum, out);
}